// GNN_77421080477842
// MI455X (gfx1250) — compile-verified
//
#include <hip/hip_runtime.h>
#include <hip/hip_bf16.h>

// GNN: 3 layers of x <- segsum(x[src] -> dst) @ W_rel + x @ W_root + b
// N=50000 nodes, E=800000 edges, D=128.

#define N_NODES 50000
#define N_EDGES 800000
#define DIM     128
#define LAYERS  3

typedef __attribute__((ext_vector_type(2))) float v2f;
typedef __attribute__((ext_vector_type(8))) float v8f;

// ---------------------------------------------------------------------------
// Scatter-add aggregation: one wave (32 lanes) per edge.
// Lane l handles columns [4l, 4l+3]: one coalesced b128 row load + 4 f32
// atomics into the destination row (L2-resident -> L2 atomic units).
// ---------------------------------------------------------------------------
__global__ __launch_bounds__(256) void gnn_scatter(
    const float* __restrict__ x,
    const int*   __restrict__ src,
    const int*   __restrict__ dst,
    float*       __restrict__ agg)
{
    const int e = blockIdx.x * 8 + (threadIdx.x >> 5);
    if (e >= N_EDGES) return;
    const int lane = threadIdx.x & 31;

    const int s = src[e];
    const int d = dst[e];

    const float4 v = reinterpret_cast<const float4*>(x + (size_t)s * DIM)[lane];
    float* p = agg + (size_t)d * DIM + lane * 4;
    unsafeAtomicAdd(p + 0, v.x);   // global_atomic_add_f32
    unsafeAtomicAdd(p + 1, v.y);
    unsafeAtomicAdd(p + 2, v.z);
    unsafeAtomicAdd(p + 3, v.w);
}

// ---------------------------------------------------------------------------
// Fused dual-GEMM + bias:
//   out[m, n] = sum_k agg[m,k]*Wrel[k,n] + sum_k x[m,k]*Wroot[k,n] + b[n]
// Block = 256 threads = 8 waves. Block owns rows [m0, m0+16); wave w owns
// columns [16w, 16w+16). K=128 swept in steps of 4 with v_wmma_f32_16x16x4_f32
// (f32 A/B/C -> exact f32 path; GEMM is not the bottleneck so no downcast).
//
// WMMA f32 16x16x4 operand layout (wave32):
//   A (16x4):  lane l -> row m = l&15, ks = 2*(l>>4); VGPR0=K[ks], VGPR1=K[ks+1]
//   B (4x16):  lane l -> col n = l&15, ks = 2*(l>>4); VGPR0=K[ks], VGPR1=K[ks+1]
//   C/D (16x16): VGPR r, lane l -> row m0 + r + 8*(l>>4), col n0 + (l&15)
// ---------------------------------------------------------------------------
#define APAD 130   // LDS row stride in floats (pad 128 -> 130: conflict-free b64)

__global__ __launch_bounds__(256) void gnn_gemm(
    const float* __restrict__ agg,
    const float* __restrict__ xin,
    const float* __restrict__ Wrel,
    const float* __restrict__ Wroot,
    const float* __restrict__ bias,
    float*       __restrict__ out)
{
    __shared__ float sAgg[16 * APAD];
    __shared__ float sX  [16 * APAD];

    const int t  = threadIdx.x;
    const int m0 = blockIdx.x * 16;          // N_NODES/16 == 3125 exact

    // Cooperative stage of the 16x128 A-panels (agg and x) into LDS.
    // 2048 floats per panel = 512 float4 -> 2 float4 per thread.
    #pragma unroll
    for (int i = 0; i < 2; ++i) {
        const int f4  = t + i * 256;         // 0..511
        const int row = f4 >> 5;             // 32 float4 per row
        const int c4  = (f4 & 31) * 4;
        const float4 va = *reinterpret_cast<const float4*>(agg + (size_t)(m0 + row) * DIM + c4);
        const float4 vx = *reinterpret_cast<const float4*>(xin + (size_t)(m0 + row) * DIM + c4);
        *reinterpret_cast<float4*>(&sAgg[row * APAD + c4]) = va;
        *reinterpret_cast<float4*>(&sX  [row * APAD + c4]) = vx;
    }
    __syncthreads();

    const int wave  = t >> 5;
    const int lane  = t & 31;
    const int n0    = wave * 16;             // 8 waves cover all 128 columns
    const int mn    = lane & 15;             // row (for A) / col (for B,C)
    const int khalf = lane >> 4;             // 0 or 1

    v8f c = {};                              // f32 accumulator tile

    for (int k = 0; k < DIM; k += 4) {
        const int ka = k + 2 * khalf;

        // A fragments from LDS (ds_load_b64, conflict-free via APAD)
        const v2f a_rel  = *reinterpret_cast<const v2f*>(&sAgg[mn * APAD + ka]);
        const v2f a_root = *reinterpret_cast<const v2f*>(&sX  [mn * APAD + ka]);

        // B fragments straight from global (64 KB weights: L2/WGP$ resident)
        v2f b_rel, b_root;
        b_rel.x  = Wrel [(size_t)(ka    ) * DIM + n0 + mn];
        b_rel.y  = Wrel [(size_t)(ka + 1) * DIM + n0 + mn];
        b_root.x = Wroot[(size_t)(ka    ) * DIM + n0 + mn];
        b_root.y = Wroot[(size_t)(ka + 1) * DIM + n0 + mn];

        c = __builtin_amdgcn_wmma_f32_16x16x4_f32(false, a_rel,  false, b_rel,
                                                  (short)0, c, false, false);
        c = __builtin_amdgcn_wmma_f32_16x16x4_f32(false, a_root, false, b_root,
                                                  (short)0, c, false, false);
    }

    const float bv = bias[n0 + mn];          // bias broadcast over rows

    #pragma unroll
    for (int r = 0; r < 8; ++r) {
        const int row = m0 + r + 8 * khalf;
        out[(size_t)row * DIM + n0 + mn] = c[r] + bv;
    }
}

// ---------------------------------------------------------------------------
// Host-side launch. d_ws layout: [buf0 | buf1 | agg], each N*D floats
// (3 * 25.6 MB = 76.8 MB). Layer chain: x -> buf0 -> buf1 -> d_out.
// ---------------------------------------------------------------------------
extern "C" void kernel_launch(void* const* d_in, const int* in_sizes, int n_in,
                              void* d_out, int out_size, void* d_ws, size_t ws_size,
                              hipStream_t stream) {
    const float* x0    = (const float*)d_in[0];               // [N, D]
    const int*   eidx  = (const int*)  d_in[1];               // [2, E]
    const float* Wrel  = (const float*)d_in[2];               // [L, D, D]
    const float* Wroot = (const float*)d_in[3];               // [L, D, D]
    const float* bias  = (const float*)d_in[4];               // [L, D]
    float*       out   = (float*)d_out;

    const size_t nodeElems = (size_t)N_NODES * DIM;
    float* buf0 = (float*)d_ws;
    float* buf1 = buf0 + nodeElems;
    float* agg  = buf1 + nodeElems;

    const int* src = eidx;
    const int* dst = eidx + N_EDGES;

    const float* cur = x0;
    float* outs[LAYERS] = {buf0, buf1, out};

    for (int l = 0; l < LAYERS; ++l) {
        hipMemsetAsync(agg, 0, nodeElems * sizeof(float), stream);

        gnn_scatter<<<(N_EDGES + 7) / 8, 256, 0, stream>>>(cur, src, dst, agg);

        gnn_gemm<<<N_NODES / 16, 256, 0, stream>>>(
            agg, cur,
            Wrel  + (size_t)l * DIM * DIM,
            Wroot + (size_t)l * DIM * DIM,
            bias  + (size_t)l * DIM,
            outs[l]);

        cur = outs[l];
    }
}